// CTCLayer_87024627351608
// MI455X (gfx1250) — compile-verified
//
#include <hip/hip_runtime.h>

// Problem constants (from reference: B=256, T=512, C=512, L=64)
#define B_    256
#define T_    512
#define C_    512
#define L_    64
#define S_    (2 * L_ + 1)   // 129 extended states
#define BLANK_ (C_ - 1)      // 511
#define NBUF  4              // async prefetch depth (power of two)
#define NEGV  (-1e30f)
#define EPSV  (1e-7f)

typedef __attribute__((ext_vector_type(4))) float v4f;

// ---------------------------------------------------------------------------
// Bandwidth kernel: copy y_pred [B,T,C] to output (second tuple element).
// 16 B per lane, non-temporal (streaming, never re-read from cache).
// ---------------------------------------------------------------------------
__global__ __launch_bounds__(256) void ypred_copy_kernel(
    const v4f* __restrict__ src, v4f* __restrict__ dst) {
  size_t i = (size_t)blockIdx.x * 256 + threadIdx.x;
  v4f v = __builtin_nontemporal_load(src + i);
  __builtin_nontemporal_store(v, dst + i);
}

// ---------------------------------------------------------------------------
// CDNA5 async global->LDS helpers (ASYNCcnt-tracked copy engine).
// GVS mode: 64-bit SGPR base + 32-bit per-lane byte offset; dst is an LDS
// byte address in a VGPR (low 32 bits of the generic shared-pointer value).
// ---------------------------------------------------------------------------
__device__ __forceinline__ void async_row_chunk_b128(const float* sbase,
                                                     unsigned goff,
                                                     unsigned lds_addr) {
  asm volatile("global_load_async_to_lds_b128 %0, %1, %2"
               :
               : "v"(lds_addr), "v"(goff), "s"(sbase)
               : "memory");
}
__device__ __forceinline__ void wait_async_le3() {
  asm volatile("s_wait_asynccnt 3" ::: "memory");
}
__device__ __forceinline__ void wait_async_le2() {
  asm volatile("s_wait_asynccnt 2" ::: "memory");
}
__device__ __forceinline__ void wait_async_le0() {
  asm volatile("s_wait_asynccnt 0" ::: "memory");
}

// 3-way logaddexp step for one alpha state (max-trick; NEGV stays ~NEGV).
__device__ __forceinline__ float ctc_step(float a, float a1, float a2,
                                          float p) {
  float m = fmaxf(a, fmaxf(a1, a2));
  float s = __expf(a - m) + __expf(a1 - m) + __expf(a2 - m);
  return m + __logf(s) + __logf(p + EPSV);
}

// ---------------------------------------------------------------------------
// CTC forward kernel: one block (256 threads = 8 waves) per batch element.
// Rows of y_pred[b] are streamed into an NBUF-deep LDS ring via the async
// copy engine while the alpha recurrence runs.
// ---------------------------------------------------------------------------
__global__ __launch_bounds__(256) void ctc_forward_kernel(
    const int* __restrict__ y_true, const float* __restrict__ y_pred,
    float* __restrict__ loss) {
  __shared__ float rowbuf[NBUF][C_];   // 8 KB ring of probability rows
  __shared__ float alpha[2][S_];
  __shared__ int ext_sh[S_];
  __shared__ int skip_sh[S_];

  const int b = blockIdx.x;
  const int tid = threadIdx.x;
  const float* rowbase = y_pred + (size_t)b * (size_t)(T_ * C_);

  // Extended label sequence: blank, l0, blank, l1, ..., blank
  if (tid < S_) {
    int e = BLANK_;
    if (tid & 1) e = y_true[b * L_ + (tid >> 1)];
    ext_sh[tid] = e;
  }
  __syncthreads();
  if (tid < S_) {
    int allow = 0;
    if (tid >= 2) {
      int e = ext_sh[tid];
      allow = (e != BLANK_) && (e != ext_sh[tid - 2]);
    }
    skip_sh[tid] = allow;
  }

  // Per-lane addresses for the async row copies (128 lanes x 16B = 2KB row).
  const unsigned lane_byte = (unsigned)(tid << 4);
  const unsigned lds_base = (unsigned)(unsigned long long)(&rowbuf[0][0]);

  // Prefetch rows 0..NBUF-1 into the ring.
  if (tid < (C_ * 4 / 16)) {
#pragma unroll
    for (int p = 0; p < NBUF; ++p) {
      async_row_chunk_b128(rowbase, (unsigned)(p * C_ * 4) + lane_byte,
                           lds_base + (unsigned)(p * C_ * 4) + lane_byte);
    }
  }
  wait_async_le3();   // row 0 chunks of this wave complete
  __syncthreads();    // => row 0 fully resident for all waves

  // alpha at t=0
  float a0 = NEGV;
  if (tid < 2) a0 = __logf(rowbuf[0][ext_sh[tid]] + EPSV);
  if (tid < S_) alpha[0][tid] = a0;
  wait_async_le2();   // row 1 complete for this wave
  __syncthreads();    // publishes alpha0 + row 1 everywhere
  int cur = 0;

  // Main pipelined scan: at step t, row t is resident; issue row t+NBUF-1.
  for (int t = 1; t <= T_ - NBUF; ++t) {
    if (tid < (C_ * 4 / 16)) {
      int nr = t + NBUF - 1;
      async_row_chunk_b128(
          rowbase, (unsigned)(nr * C_ * 4) + lane_byte,
          lds_base + (unsigned)((nr & (NBUF - 1)) * C_ * 4) + lane_byte);
    }
    const float* row = rowbuf[t & (NBUF - 1)];
    if (tid < S_) {
      float a = alpha[cur][tid];
      float a1 = (tid >= 1) ? alpha[cur][tid - 1] : NEGV;
      float a2 = skip_sh[tid] ? alpha[cur][tid - 2] : NEGV;
      alpha[cur ^ 1][tid] = ctc_step(a, a1, a2, row[ext_sh[tid]]);
    }
    wait_async_le2();  // row t+1 done (<=2 of rows t+1..t+3 may remain)
    __syncthreads();   // row t+1 visible to all waves; alpha published
    cur ^= 1;
  }

  // Drain: remaining NBUF-1 rows are all resident after a full wait.
  wait_async_le0();
  __syncthreads();
  for (int t = T_ - NBUF + 1; t < T_; ++t) {
    const float* row = rowbuf[t & (NBUF - 1)];
    if (tid < S_) {
      float a = alpha[cur][tid];
      float a1 = (tid >= 1) ? alpha[cur][tid - 1] : NEGV;
      float a2 = skip_sh[tid] ? alpha[cur][tid - 2] : NEGV;
      alpha[cur ^ 1][tid] = ctc_step(a, a1, a2, row[ext_sh[tid]]);
    }
    __syncthreads();
    cur ^= 1;
  }

  if (tid == 0) {
    float x = alpha[cur][S_ - 1];
    float y = alpha[cur][S_ - 2];
    float m = fmaxf(x, y);
    loss[b] = -(m + __logf(__expf(x - m) + __expf(y - m)));
  }
}

// ---------------------------------------------------------------------------
// Entry point. Inputs: d_in[0] = y_true [B,L] (int), d_in[1] = y_pred [B,T,C]
// (float32). Output: d_out = [loss(256 floats) | y_pred copy(B*T*C floats)].
// ---------------------------------------------------------------------------
extern "C" void kernel_launch(void* const* d_in, const int* in_sizes, int n_in,
                              void* d_out, int out_size, void* d_ws,
                              size_t ws_size, hipStream_t stream) {
  (void)in_sizes; (void)n_in; (void)d_ws; (void)ws_size; (void)out_size;
  const int* y_true = (const int*)d_in[0];
  const float* y_pred = (const float*)d_in[1];
  float* loss = (float*)d_out;
  // loss occupies 256 floats (1024 B) -> copy destination stays 16B-aligned.
  v4f* dst = (v4f*)((float*)d_out + B_);
  const v4f* src = (const v4f*)y_pred;

  ctc_forward_kernel<<<B_, 256, 0, stream>>>(y_true, y_pred, loss);

  const size_t n4 = (size_t)B_ * T_ * C_ / 4;  // 16,777,216 float4s
  ypred_copy_kernel<<<(unsigned)(n4 / 256), 256, 0, stream>>>(src, dst);
}